// TokenModel_1099511628329
// MI455X (gfx1250) — compile-verified
//
#include <hip/hip_runtime.h>
#include <hip/hip_bf16.h>
#include <math.h>

typedef __attribute__((ext_vector_type(2))) float v2f;
typedef __attribute__((ext_vector_type(8))) float v8f;

#define B_   64
#define S_   512
#define H_   768
#define A1_  100
#define A1P  112
#define C1_  300
#define C1P  304
#define NSC  4            // s-chunks for pooled partials

// workspace layout (float offsets)
#define OFF_TA      0
#define OFF_W1T     (OFF_TA + B_*S_)          // [A1P][H] transposed+padded
#define OFF_B1P     (OFF_W1T + A1P*H_)
#define OFF_W2P     (OFF_B1P + A1P)
#define OFF_SW1T    (OFF_W2P + A1P)           // [C1P][H] transposed+padded
#define OFF_SB1P    (OFF_SW1T + C1P*H_)
#define OFF_SW2P    (OFF_SB1P + C1P)
#define OFF_POOL    (OFF_SW2P + C1P)          // [B][H]
#define OFF_PPOOL   (OFF_POOL + B_*H_)        // [NSC][B][H]
#define OFF_ROWSUM  (OFF_PPOOL + NSC*B_*H_)
#define OFF_ROWMAX  (OFF_ROWSUM + B_)
#define OFF_ROWMIN  (OFF_ROWMAX + B_)
#define OFF_SLAB    (OFF_ROWMIN + B_)
#define OFF_TLP     (OFF_SLAB + B_)
#define OFF_SSE     (OFF_TLP + B_)

__device__ __forceinline__ float wave_sum32(float v) {
    v += __shfl_xor(v, 1, 32);  v += __shfl_xor(v, 2, 32);
    v += __shfl_xor(v, 4, 32);  v += __shfl_xor(v, 8, 32);
    v += __shfl_xor(v, 16, 32);
    return v;
}
__device__ __forceinline__ float wave_max32(float v) {
    v = fmaxf(v, __shfl_xor(v, 1, 32));  v = fmaxf(v, __shfl_xor(v, 2, 32));
    v = fmaxf(v, __shfl_xor(v, 4, 32));  v = fmaxf(v, __shfl_xor(v, 8, 32));
    v = fmaxf(v, __shfl_xor(v, 16, 32));
    return v;
}
__device__ __forceinline__ float wave_min32(float v) {
    v = fminf(v, __shfl_xor(v, 1, 32));  v = fminf(v, __shfl_xor(v, 2, 32));
    v = fminf(v, __shfl_xor(v, 4, 32));  v = fminf(v, __shfl_xor(v, 8, 32));
    v = fminf(v, __shfl_xor(v, 16, 32));
    return v;
}
__device__ __forceinline__ float sigmoidf_(float x) {
    return 1.0f / (1.0f + expf(-x));
}

// ---------------------------------------------------------------------------
// pad+transpose: dst[n*H + k] = (n<N) ? src[k*N + n] : 0     (dst: [Npad][K])
// ---------------------------------------------------------------------------
__global__ void padT_kernel(const float* __restrict__ src, float* __restrict__ dst,
                            int K, int N, int Npad) {
    int i = blockIdx.x * blockDim.x + threadIdx.x;
    if (i >= Npad * K) return;
    int n = i / K, k = i - n * K;
    dst[i] = (n < N) ? src[k * N + n] : 0.0f;
}

// pad vector: dst[n] = (n<N) ? src[n] : 0
__global__ void padV_kernel(const float* __restrict__ src, float* __restrict__ dst,
                            int N, int Npad) {
    int n = blockIdx.x * blockDim.x + threadIdx.x;
    if (n >= Npad) return;
    dst[n] = (n < N) ? src[n] : 0.0f;
}

// ---------------------------------------------------------------------------
// Kernel 1: token_att = sigmoid(tanh(HS @ w1 + b1) @ w2 + b2)     [B*S]
// fp32 WMMA 16x16x4. 8 waves/block; each wave owns a 16-token tile.
// A and B fragments are both contiguous-in-k v2f loads (B from transposed w1).
// ---------------------------------------------------------------------------
__global__ __launch_bounds__(256) void token_att_kernel(
    const float* __restrict__ hs,     // [B*S, H]
    const float* __restrict__ w1t,    // [A1P, H] transposed, zero-padded
    const float* __restrict__ b1p,    // [A1P]
    const float* __restrict__ w2p,    // [A1P]
    const float* __restrict__ b2,     // [1]
    float* __restrict__ token_att)    // [B*S]
{
    const int lane = threadIdx.x & 31;
    const int wave = threadIdx.x >> 5;
    const int tile = blockIdx.x * 8 + wave;          // 2048 tiles
    const int base = tile * 16;
    const int m    = lane & 15;
    const int kofs = (lane < 16) ? 0 : 2;

    v8f acc[7];
#pragma unroll
    for (int t = 0; t < 7; ++t) acc[t] = (v8f)0.0f;

    const float* arow = hs + (size_t)(base + m) * H_ + kofs;
    const float* brow[7];
#pragma unroll
    for (int t = 0; t < 7; ++t)
        brow[t] = w1t + (size_t)((lane & 15) + 16 * t) * H_ + kofs;

    for (int k = 0; k < H_; k += 4) {
        v2f a = *(const v2f*)(arow + k);
#pragma unroll
        for (int t = 0; t < 7; ++t) {
            v2f bf = *(const v2f*)(brow[t] + k);
            acc[t] = __builtin_amdgcn_wmma_f32_16x16x4_f32(
                false, a, false, bf, (short)0, acc[t], false, false);
        }
    }

    // second layer: per-lane column n = (lane&15)+16t, rows r (+8 for upper half)
    const int ncol = lane & 15;
    float w2v[7], b1v[7];
#pragma unroll
    for (int t = 0; t < 7; ++t) { w2v[t] = w2p[ncol + 16 * t]; b1v[t] = b1p[ncol + 16 * t]; }

    const float b2v = b2[0];
    float att_val = 0.0f;
#pragma unroll
    for (int r = 0; r < 8; ++r) {
        float s = 0.0f;
#pragma unroll
        for (int t = 0; t < 7; ++t) s += tanhf(acc[t][r] + b1v[t]) * w2v[t];
        // sum over the 16-lane group (rows r / r+8 split by half-wave)
        s += __shfl_xor(s, 1, 32); s += __shfl_xor(s, 2, 32);
        s += __shfl_xor(s, 4, 32); s += __shfl_xor(s, 8, 32);
        if ((lane & 15) == r) att_val = s;
    }
    if ((lane & 15) < 8) {
        int mm = (lane & 7) + ((lane >= 16) ? 8 : 0);
        token_att[base + mm] = sigmoidf_(att_val + b2v);
    }
}

// ---------------------------------------------------------------------------
// Kernel 2: per-row segment-max + masking + row reductions. 1 block / row.
// ---------------------------------------------------------------------------
__global__ __launch_bounds__(256) void row_kernel(
    const float* __restrict__ token_att,
    const float* __restrict__ labels,
    const int*   __restrict__ offmap,       // [B,S,2]
    float* __restrict__ out_masked,         // d_out+5  [B*S]
    float* __restrict__ rowsum, float* __restrict__ rowmax,
    float* __restrict__ rowmin, float* __restrict__ slab,
    float* __restrict__ tlp)
{
    __shared__ float s_att[S_];
    __shared__ float s_w[S_];
    __shared__ int   s_start[S_];
    __shared__ float red[5][8];

    const int b   = blockIdx.x;
    const int tid = threadIdx.x;

    for (int s = tid; s < S_; s += 256) {
        s_att[s]   = token_att[b * S_ + s];
        s_start[s] = (offmap[(b * S_ + s) * 2] == 0) ? 1 : 0;
        s_w[s]     = 0.0f;
    }
    __syncthreads();

    if (tid == 0) {
        // sequential segment max; first token is guaranteed a word start
        int start = 0; float mx = s_att[0];
        for (int s = 1; s < S_; ++s) {
            if (s_start[s]) { s_w[start] = mx; start = s; mx = s_att[s]; }
            else            { mx = fmaxf(mx, s_att[s]); }
        }
        s_w[start] = mx;
    }
    __syncthreads();

    float lsum = 0.0f, lmax = -INFINITY, lmin = INFINITY, llab = -INFINITY, ltl = 0.0f;
    for (int s = tid; s < S_; s += 256) {
        float lab  = labels[b * S_ + s];
        bool  mask = (lab != -1.0f) && (s_start[s] != 0);
        float mk   = mask ? s_w[s] : 0.0f;         // word_att * token_mask
        out_masked[b * S_ + s] = mk;
        lsum += mk;
        lmax  = fmaxf(lmax, mk);
        float om = (mk == 0.0f) ? 1.0f : mk;
        lmin  = fminf(lmin, om);
        llab  = fmaxf(llab, lab);
        float zl = (lab != -1.0f) ? lab : 0.0f;
        float d  = mk - zl;
        ltl += d * d;
    }
    lsum = wave_sum32(lsum);  lmax = wave_max32(lmax);
    lmin = wave_min32(lmin);  llab = wave_max32(llab);
    ltl  = wave_sum32(ltl);
    const int lane = tid & 31, wv = tid >> 5;
    if (lane == 0) { red[0][wv]=lsum; red[1][wv]=lmax; red[2][wv]=lmin; red[3][wv]=llab; red[4][wv]=ltl; }
    __syncthreads();
    if (tid == 0) {
        float a0 = 0, a1 = -INFINITY, a2 = INFINITY, a3 = -INFINITY, a4 = 0;
        for (int w = 0; w < 8; ++w) {
            a0 += red[0][w]; a1 = fmaxf(a1, red[1][w]); a2 = fminf(a2, red[2][w]);
            a3 = fmaxf(a3, red[3][w]); a4 += red[4][w];
        }
        rowsum[b] = a0; rowmax[b] = a1; rowmin[b] = a2; slab[b] = a3; tlp[b] = a4;
    }
}

// ---------------------------------------------------------------------------
// Kernel 3a: partial pooled over an S-chunk.
// grid (B, H/128, NSC), 128 threads. ppool[sc][b][h] = sum_{s in chunk} ...
// ---------------------------------------------------------------------------
__global__ __launch_bounds__(128) void pooled_partial_kernel(
    const float* __restrict__ hs,
    const float* __restrict__ masked,       // d_out+5
    const float* __restrict__ rowsum,
    float* __restrict__ ppool)              // [NSC][B][H]
{
    __shared__ float s_attn[S_ / NSC];
    const int b  = blockIdx.x;
    const int sc = blockIdx.z;
    const int s0 = sc * (S_ / NSC);
    const float inv = 1.0f / rowsum[b];
    if (threadIdx.x < S_ / NSC)
        s_attn[threadIdx.x] = masked[b * S_ + s0 + threadIdx.x] * inv;
    __syncthreads();

    const int h = blockIdx.y * 128 + threadIdx.x;
    const float* p = hs + (size_t)b * S_ * H_ + (size_t)s0 * H_ + h;
    float acc = 0.0f;
#pragma unroll 4
    for (int s = 0; s < S_ / NSC; ++s)
        acc += p[(size_t)s * H_] * s_attn[s];
    ppool[((size_t)sc * B_ + b) * H_ + h] = acc;
}

// Kernel 3b: reduce partials -> pooled [B][H]
__global__ __launch_bounds__(256) void pooled_reduce_kernel(
    const float* __restrict__ ppool, float* __restrict__ pooled) {
    int i = blockIdx.x * blockDim.x + threadIdx.x;
    if (i >= B_ * H_) return;
    float a = 0.0f;
#pragma unroll
    for (int sc = 0; sc < NSC; ++sc) a += ppool[(size_t)sc * B_ * H_ + i];
    pooled[i] = a;
}

// ---------------------------------------------------------------------------
// Kernel 4: sent = sigmoid(tanh(pooled @ sw1 + sb1) @ sw2 + sb2)   [B]
// 1 block, 4 waves; wave w owns rows 16w..16w+15. fp32 WMMA over 19 N-tiles.
// ---------------------------------------------------------------------------
__global__ __launch_bounds__(128) void sentence_kernel(
    const float* __restrict__ pooled,   // [B,H]
    const float* __restrict__ sw1t,     // [C1P,H] transposed, zero-padded
    const float* __restrict__ sb1p,     // [C1P]
    const float* __restrict__ sw2p,     // [C1P]
    const float* __restrict__ sb2,      // [1]
    const float* __restrict__ slab,     // [B]
    float* __restrict__ sent_out,       // d_out + 5 + B*S
    float* __restrict__ sse)            // [B]
{
    const int lane  = threadIdx.x & 31;
    const int wave  = threadIdx.x >> 5;
    const int wbase = wave * 16;
    const int m     = lane & 15;
    const int kofs  = (lane < 16) ? 0 : 2;

    float lsum[8];
#pragma unroll
    for (int r = 0; r < 8; ++r) lsum[r] = 0.0f;

    const float* arow = pooled + (size_t)(wbase + m) * H_ + kofs;

    for (int nt = 0; nt < C1P / 16; ++nt) {          // 19 tiles
        const int n = nt * 16 + (lane & 15);
        const float* brow = sw1t + (size_t)n * H_ + kofs;
        v8f c = (v8f)0.0f;
        for (int k = 0; k < H_; k += 4) {
            v2f a  = *(const v2f*)(arow + k);
            v2f bf = *(const v2f*)(brow + k);
            c = __builtin_amdgcn_wmma_f32_16x16x4_f32(
                false, a, false, bf, (short)0, c, false, false);
        }
        const float sb1v = sb1p[n], sw2v = sw2p[n];
#pragma unroll
        for (int r = 0; r < 8; ++r) lsum[r] += tanhf(c[r] + sb1v) * sw2v;
    }

    const float b2v = sb2[0];
    float logit = 0.0f;
#pragma unroll
    for (int r = 0; r < 8; ++r) {
        float s = lsum[r];
        s += __shfl_xor(s, 1, 32); s += __shfl_xor(s, 2, 32);
        s += __shfl_xor(s, 4, 32); s += __shfl_xor(s, 8, 32);
        if ((lane & 15) == r) logit = s;
    }
    if ((lane & 15) < 8) {
        const int bb = wbase + (lane & 7) + ((lane >= 16) ? 8 : 0);
        float sv = sigmoidf_(logit + b2v);
        sent_out[bb] = sv;
        float d = sv - slab[bb];
        sse[bb] = d * d;
    }
}

// ---------------------------------------------------------------------------
// Kernel 5: final loss assembly -> out[0..4]
// ---------------------------------------------------------------------------
__global__ __launch_bounds__(64) void final_kernel(
    const float* __restrict__ sse, const float* __restrict__ tlp,
    const float* __restrict__ rowmin, const float* __restrict__ rowmax,
    const float* __restrict__ slab, float* __restrict__ out)
{
    __shared__ float red[4][2];
    const int t = threadIdx.x;         // 0..63 == row index
    float sl = sse[t];
    float tl = tlp[t];
    float ra = rowmin[t] * rowmin[t];
    float db = rowmax[t] - slab[t];
    float rb = db * db;
    sl = wave_sum32(sl); tl = wave_sum32(tl); ra = wave_sum32(ra); rb = wave_sum32(rb);
    const int lane = t & 31, wv = t >> 5;
    if (lane == 0) { red[0][wv]=sl; red[1][wv]=tl; red[2][wv]=ra; red[3][wv]=rb; }
    __syncthreads();
    if (t == 0) {
        float SL = red[0][0] + red[0][1];
        float TL = red[1][0] + red[1][1];
        float RA = red[2][0] + red[2][1];
        float RB = red[3][0] + red[3][1];
        out[0] = 1.0f * SL + 1.0f * TL + 0.01f * (RA + RB);
        out[1] = SL; out[2] = TL; out[3] = RA; out[4] = RB;
    }
}

// ---------------------------------------------------------------------------
extern "C" void kernel_launch(void* const* d_in, const int* in_sizes, int n_in,
                              void* d_out, int out_size, void* d_ws, size_t ws_size,
                              hipStream_t stream) {
    (void)in_sizes; (void)n_in; (void)out_size; (void)ws_size;

    const float* hs     = (const float*)d_in[0];
    const float* w1     = (const float*)d_in[1];
    const float* b1     = (const float*)d_in[2];
    const float* w2     = (const float*)d_in[3];
    const float* b2     = (const float*)d_in[4];
    const float* sw1    = (const float*)d_in[5];
    const float* sb1    = (const float*)d_in[6];
    const float* sw2    = (const float*)d_in[7];
    const float* sb2    = (const float*)d_in[8];
    const float* labels = (const float*)d_in[9];
    const int*   offmap = (const int*)d_in[10];
    float* out = (float*)d_out;
    float* ws  = (float*)d_ws;

    // pad + transpose weight matrices; pad bias/output vectors
    {
        int t1 = A1P * H_;
        padT_kernel<<<(t1 + 255) / 256, 256, 0, stream>>>(w1, ws + OFF_W1T, H_, A1_, A1P);
        int t2 = C1P * H_;
        padT_kernel<<<(t2 + 255) / 256, 256, 0, stream>>>(sw1, ws + OFF_SW1T, H_, C1_, C1P);
        padV_kernel<<<1, A1P, 0, stream>>>(b1,  ws + OFF_B1P,  A1_, A1P);
        padV_kernel<<<1, A1P, 0, stream>>>(w2,  ws + OFF_W2P,  A1_, A1P);
        padV_kernel<<<(C1P + 255) / 256, 256, 0, stream>>>(sb1, ws + OFF_SB1P, C1_, C1P);
        padV_kernel<<<(C1P + 255) / 256, 256, 0, stream>>>(sw2, ws + OFF_SW2P, C1_, C1P);
    }

    token_att_kernel<<<(B_ * S_) / (16 * 8), 256, 0, stream>>>(
        hs, ws + OFF_W1T, ws + OFF_B1P, ws + OFF_W2P, b2, ws + OFF_TA);

    row_kernel<<<B_, 256, 0, stream>>>(
        ws + OFF_TA, labels, offmap, out + 5,
        ws + OFF_ROWSUM, ws + OFF_ROWMAX, ws + OFF_ROWMIN, ws + OFF_SLAB, ws + OFF_TLP);

    dim3 g3(B_, H_ / 128, NSC);
    pooled_partial_kernel<<<g3, 128, 0, stream>>>(hs, out + 5, ws + OFF_ROWSUM, ws + OFF_PPOOL);
    pooled_reduce_kernel<<<(B_ * H_ + 255) / 256, 256, 0, stream>>>(ws + OFF_PPOOL, ws + OFF_POOL);

    sentence_kernel<<<1, 128, 0, stream>>>(
        ws + OFF_POOL, ws + OFF_SW1T, ws + OFF_SB1P, ws + OFF_SW2P, sb2, ws + OFF_SLAB,
        out + 5 + B_ * S_, ws + OFF_SSE);

    final_kernel<<<1, 64, 0, stream>>>(
        ws + OFF_SSE, ws + OFF_TLP, ws + OFF_ROWMIN, ws + OFF_ROWMAX, ws + OFF_SLAB, out);
}